// MultiresHashTableEncoding_2525440770198
// MI455X (gfx1250) — compile-verified
//
#include <hip/hip_runtime.h>
#include <stdint.h>

typedef float v2f __attribute__((ext_vector_type(2)));

#define NLVL        16
#define PTS_PER_THR 4
#define PTS_PER_BLK 64   // 256 threads / 16 levels * 4 points each
#define BLK         256

#if defined(__has_builtin)
#if __has_builtin(__builtin_amdgcn_global_load_async_to_lds_b32)
#define HAVE_ASYNC_LDS 1
#endif
#endif
#ifndef HAVE_ASYNC_LDS
#define HAVE_ASYNC_LDS 0
#endif

// ---- compile-time magic-division constants (Granlund-Montgomery w/ fixup) --
// q = umulhi(n,M32); t = ((n-q)>>1)+q; quot = t>>(s-1); r = n - quot*d
// Exact for all n < 2^32 iff  d - (2^(32+s) mod d) <= 2^s,  s = ceil(log2 d).
constexpr int clog2(uint64_t d) { int k = 0; while ((uint64_t(1) << k) < d) ++k; return k; }
constexpr uint32_t magicM32(uint32_t d) {
  const int s = clog2(d);
  const uint64_t M33 = ((uint64_t(1) << (32 + s)) / d) + 1;   // in (2^32, 2^33]
  return (uint32_t)(M33 - (uint64_t(1) << 32));
}
constexpr uint32_t sdpack(uint32_t d) {   // [24:20]=s-1, [19:0]=d (d<=2^19)
  return ((uint32_t)(clog2(d) - 1) << 20) | d;
}
constexpr bool magic_ok(uint32_t d) {
  const int s = clog2(d);
  const uint64_t p2 = uint64_t(1) << (32 + s);
  const uint64_t M33 = p2 / d + 1;
  return (d - (p2 % d)) <= (uint64_t(1) << s)      // error bound
      && M33 > (uint64_t(1) << 32) && M33 <= (uint64_t(1) << 33)
      && d <= (1u << 20);
}
constexpr bool all_magic_ok() {
  const uint32_t ds[6] = {4096u, 12167u, 32768u, 91125u, 262144u, 524288u};
  for (int i = 0; i < 6; ++i) if (!magic_ok(ds[i])) return false;
  return true;
}
static_assert(all_magic_ok(), "magic division constants invalid");

// Per-level constants packed to 16B so one b128 constant load fetches all.
struct LvlC { float scale; float fmax; uint32_t M32; uint32_t sd; };

__constant__ LvlC kL[NLVL] = {
  {15.f,   14.f,   magicM32(4096u),   sdpack(4096u)},    // 16^3
  {22.f,   21.f,   magicM32(12167u),  sdpack(12167u)},   // 23^3
  {31.f,   30.f,   magicM32(32768u),  sdpack(32768u)},   // 32^3
  {44.f,   43.f,   magicM32(91125u),  sdpack(91125u)},   // 45^3
  {63.f,   62.f,   magicM32(262144u), sdpack(262144u)},  // 64^3
  {90.f,   89.f,   magicM32(524288u), sdpack(524288u)},  // 2^19 from here on
  {127.f,  126.f,  magicM32(524288u), sdpack(524288u)},
  {180.f,  179.f,  magicM32(524288u), sdpack(524288u)},
  {255.f,  254.f,  magicM32(524288u), sdpack(524288u)},
  {361.f,  360.f,  magicM32(524288u), sdpack(524288u)},
  {511.f,  510.f,  magicM32(524288u), sdpack(524288u)},
  {723.f,  722.f,  magicM32(524288u), sdpack(524288u)},
  {1023.f, 1022.f, magicM32(524288u), sdpack(524288u)},
  {1447.f, 1446.f, magicM32(524288u), sdpack(524288u)},
  {2047.f, 2046.f, magicM32(524288u), sdpack(524288u)},
  {2895.f, 2894.f, magicM32(524288u), sdpack(524288u)},
};

struct TabPtrs { const float* p[NLVL]; };

// Thread layout (wave32): lane = level + 16*point_sub. Lanes 0-15 emit the 32
// contiguous output floats of one point -> each half-wave stores one fully
// coalesced 128B line (nontemporal, to keep L2 free for the hash tables).
// Each thread handles PTS_PER_THR points at its fixed level; per-level setup
// is hoisted and up to 32 gathers per thread are in flight for the
// L2-resident (49MB < 192MB) table working set.
__global__ __launch_bounds__(BLK) void hashgrid_enc(TabPtrs tabs,
                                                    const float* __restrict__ x,
                                                    float* __restrict__ out,
                                                    int npts) {
  (void)tabs;  // read via kernarg segment below (offset 0: tabs is 1st arg)
  const int tid  = threadIdx.x;
  const int lvl  = tid & (NLVL - 1);
  const int psub = tid >> 4;

  // Stage the block's 64 points (192 floats) through LDS once, coalesced.
  __shared__ float xs[PTS_PER_BLK * 3];
  if (tid < PTS_PER_BLK * 3) {
    const int g = blockIdx.x * (PTS_PER_BLK * 3) + tid;
#if HAVE_ASYNC_LDS
    if (g < npts * 3) {
      // gfx1250 async global->LDS copy (ASYNCcnt-tracked, no VGPR staging).
      typedef __attribute__((address_space(1))) int GI;
      typedef __attribute__((address_space(3))) int LI;
      GI* src = (GI*)(uintptr_t)(x + g);
      LI* dst = (LI*)(uint32_t)(uintptr_t)&xs[tid];  // low 32b = LDS offset
      __builtin_amdgcn_global_load_async_to_lds_b32(src, dst, 0, 0);
    } else {
      xs[tid] = 0.f;
    }
#else
    xs[tid] = (g < npts * 3) ? x[g] : 0.f;
#endif
  }
#if HAVE_ASYNC_LDS
  asm volatile("s_wait_asynccnt 0x0" ::: "memory");
#endif
  __syncthreads();

  // ---- level-invariant setup (amortized over PTS_PER_THR points) ----
  const LvlC c = kL[lvl];
  const uint32_t M   = c.M32;
  const uint32_t sm1 = c.sd >> 20;
  const uint32_t d   = c.sd & 0xFFFFFu;

  // Table base pointer: TabPtrs is the FIRST kernel argument, so its 16
  // pointers live at kernarg offset 0 (confirmed: x/out s_load at 0x80).
  typedef const __attribute__((address_space(4))) unsigned long long* KAP;
  KAP ka = (KAP)__builtin_amdgcn_kernarg_segment_ptr();
  const v2f* __restrict__ tab = (const v2f*)(uintptr_t)ka[lvl];

#pragma unroll
  for (int it = 0; it < PTS_PER_THR; ++it) {
    const int ps  = psub + it * (BLK / NLVL);
    const int pid = blockIdx.x * PTS_PER_BLK + ps;
    if (pid >= npts) continue;

    const float px = fminf(fmaxf(xs[ps * 3 + 0], 0.f), 1.f);
    const float py = fminf(fmaxf(xs[ps * 3 + 1], 0.f), 1.f);
    const float pz = fminf(fmaxf(xs[ps * 3 + 2], 0.f), 1.f);

    const float gx = c.scale * px, gy = c.scale * py, gz = c.scale * pz;
    const float flx = fminf(floorf(gx), c.fmax);
    const float fly = fminf(floorf(gy), c.fmax);
    const float flz = fminf(floorf(gz), c.fmax);
    const float cfx = gx - flx, cfy = gy - fly, cfz = gz - flz;

    const uint32_t ux = (uint32_t)flx, uy = (uint32_t)fly, uz = (uint32_t)flz;
    // hash = cx ^ (19349663*cy) ^ (83492791*cz); hoist the 4 multiplies.
    const uint32_t ax0 = ux,             ax1 = ux + 1u;
    const uint32_t ay0 = 19349663u * uy, ay1 = ay0 + 19349663u;
    const uint32_t az0 = 83492791u * uz, az1 = az0 + 83492791u;

    // corner index bit2 = x-offset, bit1 = y-offset, bit0 = z-offset
    uint32_t h[8] = { ax0^ay0^az0, ax0^ay0^az1, ax0^ay1^az0, ax0^ay1^az1,
                      ax1^ay0^az0, ax1^ay0^az1, ax1^ay1^az0, ax1^ay1^az1 };

    // Branchless universal modulo: one path for all 16 levels (no exec-mask
    // divergence; levels share the wave).
    uint32_t idx[8];
#pragma unroll
    for (int i = 0; i < 8; ++i) {
      const uint32_t q    = __umulhi(h[i], M);
      const uint32_t t    = ((h[i] - q) >> 1) + q;
      const uint32_t quot = t >> sm1;
      idx[i] = h[i] - quot * d;
    }

    // 8 independent b64 gathers (L2-resident tables); issue all before use.
    v2f v[8];
#pragma unroll
    for (int i = 0; i < 8; ++i) v[i] = tab[idx[i]];

    const float wx1 = cfx, wx0 = 1.f - cfx;
    const float wy1 = cfy, wy0 = 1.f - cfy;
    const float wz1 = cfz, wz0 = 1.f - cfz;
    const float w00 = wy0 * wz0, w01 = wy0 * wz1, w10 = wy1 * wz0, w11 = wy1 * wz1;

    float f0 = wx0 * (w00*v[0].x + w01*v[1].x + w10*v[2].x + w11*v[3].x)
             + wx1 * (w00*v[4].x + w01*v[5].x + w10*v[6].x + w11*v[7].x);
    float f1 = wx0 * (w00*v[0].y + w01*v[1].y + w10*v[2].y + w11*v[3].y)
             + wx1 * (w00*v[4].y + w01*v[5].y + w10*v[6].y + w11*v[7].y);

    v2f r; r.x = f0; r.y = f1;
    // Nontemporal: stream 256MB of output past L2 so the 49MB of hash tables
    // stay resident (CDNA5 TH=NT store hint).
    __builtin_nontemporal_store(r, (v2f*)(out + (size_t)pid * (NLVL * 2) + lvl * 2));
  }
}

extern "C" void kernel_launch(void* const* d_in, const int* in_sizes, int n_in,
                              void* d_out, int out_size, void* d_ws, size_t ws_size,
                              hipStream_t stream) {
  (void)n_in; (void)out_size; (void)d_ws; (void)ws_size;
  const float* x = (const float*)d_in[0];
  const int npts = in_sizes[0] / 3;

  TabPtrs tabs;
  for (int i = 0; i < NLVL; ++i) tabs.p[i] = (const float*)d_in[1 + i];

  float* out = (float*)d_out;
  const int blocks = (npts + PTS_PER_BLK - 1) / PTS_PER_BLK;
  hipLaunchKernelGGL(hashgrid_enc, dim3(blocks), dim3(BLK), 0, stream,
                     tabs, x, out, npts);
}